// Model_88476326297683
// MI455X (gfx1250) — compile-verified
//
#include <hip/hip_runtime.h>
#include <hip/hip_bf16.h>
#include <stdint.h>

// Problem constants (from reference)
#define Dn   200
#define Cn   8
#define Vn   50000
#define NGn  20000
#define NLVn 20000
#define NLGn 5000
#define CHUNK 256
#define NCHUNK ((Vn + CHUNK - 1) / CHUNK)   // 196
#define NTAB 4

// log(0.1) + 0.5*log(2*pi)
#define LOG_NORM_CONST (-1.3836465597893728f)
#define HALF_LOG_2PI   (0.9189385332046727f)

typedef unsigned int uint32x4 __attribute__((ext_vector_type(4)));
typedef int          int32x8  __attribute__((ext_vector_type(8)));
typedef int          int32x4  __attribute__((ext_vector_type(4)));

// lgamma for x > 0: recurrence (as a product -> single log) up to >= 8,
// then 3-term Stirling. Masked recurrence iterations cost one v_mul, not a log.
__device__ __forceinline__ float lgamma_pos(float x) {
    float p = 1.0f;
    while (x < 8.0f) { p *= x; x += 1.0f; }
    float inv  = 1.0f / x;
    float inv2 = inv * inv;
    float ser  = inv * (8.3333333333e-2f + inv2 * (-2.7777777778e-3f + inv2 * 7.9365079365e-4f));
    return (x - 0.5f) * __logf(x) - x + HALF_LOG_2PI + ser - __logf(p);
}

__global__ void init_out(float* out) { out[0] = 0.0f; }

// Precompute per-(c,v) tables into ws (4 tables of C*V floats, [c][v] layout):
//   t0: fc          t1: baseline_log - log(r+eps)   t2: r   t3: lgamma(r)
// and accumulate the prior term into out.
__global__ void precompute_tables(const float* __restrict__ fc_weight,
                                  const float* __restrict__ baseline_log,
                                  const float* __restrict__ dispersion_log,
                                  const int*   __restrict__ vxg_ixs,
                                  const int*   __restrict__ vxg_to_gene,
                                  float* __restrict__ ws, float* __restrict__ out) {
    int v = blockIdx.x * blockDim.x + threadIdx.x;
    float* t_f  = ws + 0 * Cn * Vn;
    float* t_a  = ws + 1 * Cn * Vn;
    float* t_r  = ws + 2 * Cn * Vn;
    float* t_gr = ws + 3 * Cn * Vn;

    float prior = 0.0f;
    if (v < Vn) {
        int ix = vxg_ixs[v];
        int g  = vxg_to_gene[v];
#pragma unroll
        for (int c = 0; c < Cn; ++c) {
            float fc   = fc_weight[ix * Cn + c];
            float bl   = baseline_log[c * NGn + g];
            float dl   = dispersion_log[c * NGn + g];
            float disp = fminf(__expf(dl), 20.0f);
            float r    = 1.0f / disp;
            t_f [c * Vn + v] = fc;
            t_a [c * Vn + v] = bl - __logf(r + 1e-8f);   // fold -log(r+eps) into base
            t_r [c * Vn + v] = r;
            t_gr[c * Vn + v] = lgamma_pos(r);
            prior += fc * fc * 50.0f + LOG_NORM_CONST;   // 1/(2*0.01) = 50
        }
    }
    // wave32 reduction, lane 0 of each wave does one atomic
    for (int off = 16; off > 0; off >>= 1) prior += __shfl_down(prior, off);
    if ((threadIdx.x & 31) == 0) atomicAdd(out, prior);
}

// Main ELBO kernel: blockIdx.x = v-chunk, blockIdx.y = d.
__global__ void __launch_bounds__(CHUNK)
elbo_main(const float* __restrict__ lib,            // (D,C)
          const float* __restrict__ genotypes,      // (D,NLV)
          const float* __restrict__ expression_obs, // (D,C,NLG)
          const int*   __restrict__ selector,       // (V)
          const int*   __restrict__ vxg_to_lg,      // (V)
          const float* __restrict__ ws,             // 4 tables [c][v]
          float* __restrict__ out) {
    __shared__ float s_tab[NTAB][Cn * CHUNK];  // 32 KB, filled by TDM
    __shared__ float s_llib[Cn];               // log(lib[d,c])
    __shared__ float s_lgfact[128];            // log(k!) = lgamma(1+k)
    __shared__ float s_part[8];

    const int tid = threadIdx.x;
    const int d   = blockIdx.y;
    const unsigned v0 = blockIdx.x * CHUNK;
    const unsigned remaining = Vn - v0;     // grid guarantees > 0; TDM zero-fills OOB

    if (tid < Cn)   s_llib[tid]   = __logf(lib[d * Cn + tid]);
    if (tid < 128)  s_lgfact[tid] = lgamma_pos(1.0f + (float)tid);

    // --- Tensor Data Mover: DMA four 8x256 f32 tiles (row stride V) into LDS ---
    if (tid < 32) {  // wave 0 issues; TDM ignores EXEC, tracked by TENSORcnt
        unsigned lds_base = (unsigned)(uintptr_t)(&s_tab[0][0]);
#pragma unroll
        for (int t = 0; t < NTAB; ++t) {
            unsigned long long gaddr =
                (unsigned long long)(uintptr_t)(ws + (size_t)t * Cn * Vn + v0);
            uint32x4 g0;
            g0.x = 1u;                                   // count=1 valid descriptor
            g0.y = lds_base + (unsigned)t * Cn * CHUNK * 4u;  // lds_addr (bytes)
            g0.z = (unsigned)(gaddr & 0xFFFFFFFFull);    // global_addr[31:0]
            g0.w = (unsigned)((gaddr >> 32) & 0x1FFFFFFull) | (2u << 30); // addr[56:32] | type=2
            int32x8 g1;
            g1[0] = (int)(2u << 16);                     // data_size = 4 bytes
            g1[1] = (int)((remaining & 0xFFFFu) << 16);  // tensor_dim0[15:0] (OOB -> 0)
            g1[2] = (int)((remaining >> 16) | ((unsigned)Cn << 16)); // dim0 hi | tensor_dim1=8
            g1[3] = (int)((unsigned)CHUNK << 16);        // tile_dim0 = 256
            g1[4] = Cn;                                  // tile_dim1 = 8, tile_dim2 = 0
            g1[5] = Vn;                                  // tensor_dim0_stride lo32
            g1[6] = 0;
            g1[7] = 0;
            int32x4 gz  = {0, 0, 0, 0};
            int32x8 gz8 = {0, 0, 0, 0, 0, 0, 0, 0};
            __builtin_amdgcn_tensor_load_to_lds(g0, g1, gz, gz, gz8, 0);
        }
        __builtin_amdgcn_s_wait_tensorcnt(0);
    }
    __syncthreads();

    const unsigned v = v0 + tid;
    float acc = 0.0f;
    if (v < Vn) {
        const int   sel = selector[v];
        const int   lg  = vxg_to_lg[v];
        const float g   = genotypes[d * NLVn + sel];
#pragma unroll
        for (int c = 0; c < Cn; ++c) {
            float f   = s_tab[0][c * CHUNK + tid];
            float a2  = s_tab[1][c * CHUNK + tid];   // base - log(r+eps)
            float r   = s_tab[2][c * CHUNK + tid];
            float gr  = s_tab[3][c * CHUNK + tid];
            float obs = expression_obs[(size_t)(d * Cn + c) * NLGn + lg];

            // logits = log(mu+eps) - log(r+eps) ~= base + g*f + log(lib) - log(r+eps)
            float logits = fmaf(g, f, a2) + s_llib[c];
            // softplus(logits); log_sigmoid(-x) = -sp, log_sigmoid(x) = x - sp
            float sp = (logits > 15.0f) ? logits : __logf(1.0f + __expf(logits));
            float ro = r + obs;
            // r*(-sp) + obs*(logits - sp) = obs*logits - (r+obs)*sp
            float lp = fmaf(obs, logits, -ro * sp)
                     + lgamma_pos(ro) - s_lgfact[(int)obs] - gr;
            acc += lp;
        }
    }

    // block reduction: wave32 shuffles then 8 partials in LDS
    for (int off = 16; off > 0; off >>= 1) acc += __shfl_down(acc, off);
    if ((tid & 31) == 0) s_part[tid >> 5] = acc;
    __syncthreads();
    if (tid == 0) {
        float s = 0.0f;
#pragma unroll
        for (int w = 0; w < 8; ++w) s += s_part[w];
        atomicAdd(out, -s);   // elbo = -sum(log_prob) + prior
    }
}

extern "C" void kernel_launch(void* const* d_in, const int* in_sizes, int n_in,
                              void* d_out, int out_size, void* d_ws, size_t ws_size,
                              hipStream_t stream) {
    const float* fc_weight      = (const float*)d_in[0];
    const float* baseline_log   = (const float*)d_in[1];
    const float* dispersion_log = (const float*)d_in[2];
    const float* lib            = (const float*)d_in[3];
    const float* genotypes      = (const float*)d_in[4];
    const float* expression_obs = (const float*)d_in[5];
    const int*   vxg_ixs        = (const int*)d_in[6];
    const int*   vxg_to_gene    = (const int*)d_in[7];
    const int*   selector       = (const int*)d_in[8];
    const int*   vxg_to_lg      = (const int*)d_in[9];
    float* out = (float*)d_out;
    float* ws  = (float*)d_ws;   // needs 4*C*V*4 = 6.4 MB

    init_out<<<1, 1, 0, stream>>>(out);
    precompute_tables<<<NCHUNK, CHUNK, 0, stream>>>(fc_weight, baseline_log,
                                                    dispersion_log, vxg_ixs,
                                                    vxg_to_gene, ws, out);
    dim3 grid(NCHUNK, Dn);
    elbo_main<<<grid, CHUNK, 0, stream>>>(lib, genotypes, expression_obs,
                                          selector, vxg_to_lg, ws, out);
}